// DecodeDetections_12747462934797
// MI455X (gfx1250) — compile-verified
//
#include <hip/hip_runtime.h>
#include <hip/hip_bf16.h>
#include <stdint.h>

// ---------------- problem constants (match reference) ----------------
#define BATCH   32
#define NBOX    8732
#define NDIM    93          // N_CLASSES + 12
#define NCLS    80          // foreground classes (class 0 = background skipped)
#define KCAND   512         // NMS_CAND
#define NMSMAX  400         // nms_max_output_size
#define TOPK    200
#define MFLAT   (NCLS * NMSMAX)   // 32000 padded rows per batch
#define CONF_T  0.01f
#define IOU_T   0.45f
#define IMG_W   512.0f
#define IMG_H   512.0f

typedef unsigned long long u64;
typedef unsigned int       u32;

typedef unsigned int v4u __attribute__((ext_vector_type(4)));
typedef int          v8i __attribute__((ext_vector_type(8)));
typedef int          v4i __attribute__((ext_vector_type(4)));

// key packing: score descending, then original index ascending (stable-sort tie rule)
__device__ __forceinline__ u64 pack_key(float s, u32 idx) {
    if (!(s > CONF_T)) return 0ULL;
    u32 sb = __float_as_uint(s);                 // s in (0.01,1] -> positive float bits, monotone as uint
    return ((u64)sb << 32) | (u64)(0xFFFFFFFFu - idx);
}

// ---------------- TDM helper: 2D tile (tile_d0 x tile_d1) global -> LDS ----------------
// D# packing per CDNA5 ISA.
// elem_code: 0=1B 1=2B 2=4B 3=8B.  stride0 in elements.  lds_off = LDS byte offset.
__device__ __forceinline__ void tdm_load_2d(u32 lds_off, const void* gptr, u32 elem_code,
                                            u32 tensor_d0, u32 tile_d0,
                                            u32 tensor_d1, u32 tile_d1, u32 stride0) {
    u64 gaddr = (u64)gptr;
    v4u g0 = { 1u,                                     // count=1, is_restore=0, gather=0
               lds_off,                                // lds_addr (bytes)
               (u32)(gaddr & 0xFFFFFFFFu),
               (u32)(((gaddr >> 32) & 0x1FFFFFFu) | (2u << 30)) };   // addr[56:32] | type=2
    v8i g1 = { (int)(elem_code << 16),                                      // data_size
               (int)((tensor_d0 & 0xFFFFu) << 16),                          // tensor_dim0[15:0]
               (int)(((tensor_d0 >> 16) & 0xFFFFu) | ((tensor_d1 & 0xFFFFu) << 16)),
               (int)(((tensor_d1 >> 16) & 0xFFFFu) | ((tile_d0 & 0xFFFFu) << 16)),
               (int)(tile_d1 & 0xFFFFu),                                    // tile_dim1 | tile_dim2=0
               (int)stride0,                                                // tensor_dim0_stride[31:0]
               0, 0 };
    v4i g2 = { 0, 0, 0, 0 };
    v4i g3 = { 0, 0, 0, 0 };
#if defined(__clang_major__) && (__clang_major__ >= 23)
    v8i gx = { 0, 0, 0, 0, 0, 0, 0, 0 };
    __builtin_amdgcn_tensor_load_to_lds(g0, g1, g2, g3, gx, 0);
#else
    __builtin_amdgcn_tensor_load_to_lds(g0, g1, g2, g3, 0);
#endif
}

// ---------------- bitonic networks (block-wide, LDS) ----------------
template <int SIZE>
__device__ void bitonic_sort_desc(u64* a) {
    for (int k = 2; k <= SIZE; k <<= 1) {
        for (int j = k >> 1; j > 0; j >>= 1) {
            __syncthreads();
            for (int i = threadIdx.x; i < SIZE; i += blockDim.x) {
                int l = i ^ j;
                if (l > i) {
                    u64 x = a[i], y = a[l];
                    bool desc = ((i & k) == 0);
                    if (desc ? (x < y) : (x > y)) { a[i] = y; a[l] = x; }
                }
            }
        }
    }
    __syncthreads();
}

template <int SIZE>
__device__ void bitonic_merge_desc(u64* a) {   // input bitonic -> sorted descending
    for (int j = SIZE >> 1; j > 0; j >>= 1) {
        __syncthreads();
        for (int i = threadIdx.x; i < SIZE; i += blockDim.x) {
            int l = i ^ j;
            if (l > i) {
                u64 x = a[i], y = a[l];
                if (x < y) { a[i] = y; a[l] = x; }
            }
        }
    }
    __syncthreads();
}

// ---------------- kernel 1: decode boxes (streaming, HBM-bound) ----------------
__global__ __launch_bounds__(256)
void dd_decode_kernel(const float* __restrict__ y, float* __restrict__ boxes) {
    int i = blockIdx.x * blockDim.x + threadIdx.x;
    if (i >= BATCH * NBOX) return;
    const float* p = y + (size_t)i * NDIM;
    __builtin_prefetch(p + 96 * NDIM, 0, 1);     // global_prefetch_b8
    float aw = p[87], ah = p[88];
    float cx = p[81] * p[89] * aw + p[85];
    float cy = p[82] * p[90] * ah + p[86];
    float w  = expf(p[83] * p[91]) * aw;
    float h  = expf(p[84] * p[92]) * ah;
    float* o = boxes + (size_t)i * 4;
    o[0] = (cx - 0.5f * w) * IMG_W;
    o[1] = (cy - 0.5f * h) * IMG_H;
    o[2] = (cx + 0.5f * w) * IMG_W;
    o[3] = (cy + 0.5f * h) * IMG_H;
}

// ---------------- kernel 2: per-(b,c) top-512 candidates ----------------
// TDM gathers the strided score column y[b, :, 1+c] (stride 93 floats) into LDS;
// the 104MB y_pred tensor is L2-resident (192MB), so the 80 per-class passes hit L2.
struct K2Shared {
    float sc[NBOX];     // at LDS offset 0: TDM target
    u64   cur[KCAND];
    u64   chk[KCAND];
};

__global__ __launch_bounds__(256)
void dd_topk_cand_kernel(const float* __restrict__ y,
                         const float* __restrict__ boxes,
                         u64* __restrict__ cand_keys,
                         float* __restrict__ cand_boxes) {
    __shared__ K2Shared sm;
    const int b = blockIdx.x / NCLS;
    const int c = blockIdx.x % NCLS;

    if (threadIdx.x == 0) {
        tdm_load_2d(0u, y + (size_t)b * NBOX * NDIM + 1 + c,
                    /*4B*/2u, /*tensor_d0*/1u, /*tile_d0*/1u,
                    /*tensor_d1*/NBOX, /*tile_d1*/NBOX, /*stride0*/NDIM);
        __builtin_amdgcn_s_wait_tensorcnt((short)0);
    }
    // running top-512 list: all-zero keys == trivially sorted descending
    for (int i = threadIdx.x; i < KCAND; i += blockDim.x) sm.cur[i] = 0ULL;
    __syncthreads();

    for (int base = 0; base < NBOX; base += KCAND) {
        for (int i = threadIdx.x; i < KCAND; i += blockDim.x) {
            int g = base + i;
            sm.chk[i] = (g < NBOX) ? pack_key(sm.sc[g], (u32)g) : 0ULL;
        }
        bitonic_sort_desc<KCAND>(sm.chk);
        for (int i = threadIdx.x; i < KCAND; i += blockDim.x) {
            u64 a = sm.cur[i], d = sm.chk[KCAND - 1 - i];
            sm.cur[i] = (a > d) ? a : d;      // elementwise max -> bitonic, top-K superset
        }
        bitonic_merge_desc<KCAND>(sm.cur);
    }

    const size_t cbase = (size_t)blockIdx.x * KCAND;
    for (int i = threadIdx.x; i < KCAND; i += blockDim.x) {
        u64 k = sm.cur[i];
        cand_keys[cbase + i] = k;
        float bx0 = 0.f, bx1 = 0.f, bx2 = 0.f, bx3 = 0.f;
        if (k) {
            u32 idx = 0xFFFFFFFFu - (u32)(k & 0xFFFFFFFFu);
            const float* bp = boxes + ((size_t)b * NBOX + idx) * 4;
            bx0 = bp[0]; bx1 = bp[1]; bx2 = bp[2]; bx3 = bp[3];
        }
        float* op = cand_boxes + (cbase + i) * 4;
        op[0] = bx0; op[1] = bx1; op[2] = bx2; op[3] = bx3;
    }
}

// ---------------- kernel 3: greedy NMS per (b,c) ----------------
// Keys (4KB) + boxes (8KB) staged into LDS by two TDM tiles.
struct K3Shared {
    u64   keys[KCAND];        // LDS offset 0
    float bx[KCAND * 4];      // LDS offset 4096
    unsigned char sup[KCAND];
    int cnt;
};

__global__ __launch_bounds__(256)
void dd_nms_kernel(const u64* __restrict__ cand_keys,
                   const float* __restrict__ cand_boxes,
                   u64* __restrict__ kept_keys,
                   int* __restrict__ kept_slot) {
    __shared__ K3Shared sm;
    const int b = blockIdx.x / NCLS;
    const int c = blockIdx.x % NCLS;
    const size_t cbase = (size_t)blockIdx.x * KCAND;
    const size_t kbase = (size_t)b * MFLAT + (size_t)c * NMSMAX;

    if (threadIdx.x == 0) {
        tdm_load_2d((u32)__builtin_offsetof(K3Shared, keys), cand_keys + cbase,
                    /*8B*/3u, KCAND, KCAND, 1u, 1u, KCAND);
        tdm_load_2d((u32)__builtin_offsetof(K3Shared, bx), cand_boxes + cbase * 4,
                    /*4B*/2u, KCAND * 4, KCAND * 4, 1u, 1u, KCAND * 4);
        __builtin_amdgcn_s_wait_tensorcnt((short)0);
        sm.cnt = 0;
    }
    // zero this block's slice of the padded output tables
    for (int i = threadIdx.x; i < NMSMAX; i += blockDim.x) {
        kept_keys[kbase + i] = 0ULL;
        kept_slot[kbase + i] = 0;
    }
    for (int i = threadIdx.x; i < KCAND; i += blockDim.x) sm.sup[i] = 0;
    __syncthreads();

    for (int i = 0; i < KCAND; i++) {
        int  cc   = sm.cnt;
        u64  ki   = sm.keys[i];
        bool keep = (ki != 0ULL) && !sm.sup[i] && (cc < NMSMAX);   // uniform across block
        __syncthreads();                                           // all reads before writes
        if (keep) {
            float x1 = sm.bx[i * 4], y1 = sm.bx[i * 4 + 1];
            float x2 = sm.bx[i * 4 + 2], y2 = sm.bx[i * 4 + 3];
            float areaA = (x2 - x1) * (y2 - y1);
            for (int j = i + 1 + threadIdx.x; j < KCAND; j += blockDim.x) {
                float xb1 = sm.bx[j * 4], yb1 = sm.bx[j * 4 + 1];
                float xb2 = sm.bx[j * 4 + 2], yb2 = sm.bx[j * 4 + 3];
                float ix1 = fmaxf(x1, xb1), iy1 = fmaxf(y1, yb1);
                float ix2 = fminf(x2, xb2), iy2 = fminf(y2, yb2);
                float inter = fmaxf(ix2 - ix1, 0.f) * fmaxf(iy2 - iy1, 0.f);
                float areaB = (xb2 - xb1) * (yb2 - yb1);
                float uni = areaA + areaB - inter;
                float iou = (uni > 0.f) ? (inter / uni) : 0.f;
                if (iou > IOU_T) sm.sup[j] = 1;
            }
            if (threadIdx.x == 0) {
                u32 flat = (u32)(c * NMSMAX + cc);
                kept_keys[(size_t)b * MFLAT + flat] =
                    ((ki >> 32) << 32) | (u64)(0xFFFFFFFFu - flat);
                kept_slot[(size_t)b * MFLAT + flat] = i;
                sm.cnt = cc + 1;
            }
        }
        __syncthreads();
    }
}

// ---------------- kernel 4: per-batch top-200 over 32000 padded rows ----------------
// Double-buffered TDM pipeline: DMA chunk n+1 into buf[1-p] while sorting buf[p].
#define KFIN 256
struct K4Shared {
    u64 cur[KFIN];          // LDS offset 0
    u64 buf[2][KFIN];       // LDS offsets 2048, 4096
};

__global__ __launch_bounds__(256)
void dd_final_topk_kernel(const u64* __restrict__ kept_keys,
                          const int* __restrict__ kept_slot,
                          const float* __restrict__ cand_boxes,
                          float* __restrict__ out) {
    __shared__ K4Shared sm;
    const int b = blockIdx.x;
    const size_t kb = (size_t)b * MFLAT;
    const u32 buf_off0 = (u32)__builtin_offsetof(K4Shared, buf);

    for (int i = threadIdx.x; i < KFIN; i += blockDim.x) sm.cur[i] = 0ULL;  // sorted-desc identity
    if (threadIdx.x == 0)   // prologue: DMA chunk 0 -> buf[0]
        tdm_load_2d(buf_off0, kept_keys + kb, /*8B*/3u, KFIN, KFIN, 1u, 1u, KFIN);

    int p = 0;
    for (int base = 0; base < MFLAT; base += KFIN) {
        if (threadIdx.x == 0) __builtin_amdgcn_s_wait_tensorcnt((short)0);
        __syncthreads();                       // buf[p] visible to all waves
        if (threadIdx.x == 0 && base + KFIN < MFLAT)   // DMA next chunk while we sort
            tdm_load_2d(buf_off0 + (u32)((1 - p) * KFIN * sizeof(u64)),
                        kept_keys + kb + base + KFIN, 3u, KFIN, KFIN, 1u, 1u, KFIN);
        u64* chk = sm.buf[p];
        bitonic_sort_desc<KFIN>(chk);
        for (int i = threadIdx.x; i < KFIN; i += blockDim.x) {
            u64 a = sm.cur[i], d = chk[KFIN - 1 - i];
            sm.cur[i] = (a > d) ? a : d;
        }
        bitonic_merge_desc<KFIN>(sm.cur);
        p ^= 1;
    }

    for (int r = threadIdx.x; r < TOPK; r += blockDim.x) {
        float* o = out + ((size_t)b * TOPK + r) * 6;
        u64 k = sm.cur[r];
        if (k == 0ULL) {
            o[0] = 0.f; o[1] = 0.f; o[2] = 0.f; o[3] = 0.f; o[4] = 0.f; o[5] = 0.f;
        } else {
            float conf = __uint_as_float((u32)(k >> 32));
            u32 flat = 0xFFFFFFFFu - (u32)(k & 0xFFFFFFFFu);
            int c = (int)(flat / NMSMAX);
            int slot = kept_slot[kb + flat];
            const float* bp = cand_boxes + (((size_t)b * NCLS + c) * KCAND + slot) * 4;
            o[0] = (float)(c + 1);
            o[1] = conf;
            o[2] = bp[0]; o[3] = bp[1]; o[4] = bp[2]; o[5] = bp[3];
        }
    }
}

// ---------------- launch ----------------
extern "C" void kernel_launch(void* const* d_in, const int* in_sizes, int n_in,
                              void* d_out, int out_size, void* d_ws, size_t ws_size,
                              hipStream_t stream) {
    (void)in_sizes; (void)n_in; (void)out_size; (void)ws_size;
    const float* y = (const float*)d_in[0];
    float* out = (float*)d_out;
    char* ws = (char*)d_ws;

    size_t off = 0;
    float* boxes      = (float*)(ws + off); off += (size_t)BATCH * NBOX * 4 * sizeof(float);
    u64*   cand_keys  = (u64*)  (ws + off); off += (size_t)BATCH * NCLS * KCAND * sizeof(u64);
    float* cand_boxes = (float*)(ws + off); off += (size_t)BATCH * NCLS * KCAND * 4 * sizeof(float);
    u64*   kept_keys  = (u64*)  (ws + off); off += (size_t)BATCH * MFLAT * sizeof(u64);
    int*   kept_slot  = (int*)  (ws + off); off += (size_t)BATCH * MFLAT * sizeof(int);

    dd_decode_kernel<<<(BATCH * NBOX + 255) / 256, 256, 0, stream>>>(y, boxes);
    dd_topk_cand_kernel<<<BATCH * NCLS, 256, 0, stream>>>(y, boxes, cand_keys, cand_boxes);
    dd_nms_kernel<<<BATCH * NCLS, 256, 0, stream>>>(cand_keys, cand_boxes, kept_keys, kept_slot);
    dd_final_topk_kernel<<<BATCH, 256, 0, stream>>>(kept_keys, kept_slot, cand_boxes, out);
}